// PatchGCN_10625749090912
// MI455X (gfx1250) — compile-verified
//
#include <hip/hip_runtime.h>
#include <cstddef>

#define LEAKY_SLOPE 0.01f

typedef float v2f __attribute__((ext_vector_type(2)));
typedef float v8f __attribute__((ext_vector_type(8)));

__device__ __forceinline__ float lrelu(float v) { return v >= 0.0f ? v : LEAKY_SLOPE * v; }

// ---------------------------------------------------------------------------
// Degree accumulation: exact (integer-valued float adds are order-invariant)
// ---------------------------------------------------------------------------
__global__ void deg_accum(const int* __restrict__ src, const int* __restrict__ dst,
                          float* __restrict__ odeg, float* __restrict__ ideg, int e) {
    int i = blockIdx.x * blockDim.x + threadIdx.x;
    if (i < e) {
        atomicAdd(&odeg[src[i]], 1.0f);
        atomicAdd(&ideg[dst[i]], 1.0f);
    }
}

__global__ void deg_to_rsqrt(float* __restrict__ odeg, float* __restrict__ ideg, int n) {
    int i = blockIdx.x * blockDim.x + threadIdx.x;
    if (i < n) {
        odeg[i] = rsqrtf(fmaxf(odeg[i], 1.0f));
        ideg[i] = rsqrtf(fmaxf(ideg[i], 1.0f));
    }
}

// ---------------------------------------------------------------------------
// Fused GEMM:  H = pre(A) @ W
//   PRE=false: pre(A)[r,k] = A[r,k] * rs_out[r]                      (layer 0)
//   PRE=true : pre(A)[r,k] = lrelu(A[r,k]*rs_in[r] + bias[k]) * rs_out[r]
// One wave computes a 16x16 tile via V_WMMA_F32_16X16X4_F32 over K.
// Block = 8 waves sharing RT 16-row A-panels staged in LDS (pad 4 floats
// per row -> conflict-free column reads, 16B-aligned float4 stores).
// ---------------------------------------------------------------------------
template <int FIN, int FOUT, bool PRE>
__global__ void __launch_bounds__(256) gcn_gemm_wmma(
        const float* __restrict__ A, const float* __restrict__ rs_out,
        const float* __restrict__ rs_in, const float* __restrict__ bias,
        const float* __restrict__ W, float* __restrict__ H, int n) {
    constexpr int CT  = FOUT / 16;   // column tiles per row tile
    constexpr int RT  = 8 / CT;      // row tiles handled per block (8 waves)
    constexpr int LDK = FIN + 4;     // padded LDS row stride
    __shared__ float lds[RT * 16 * LDK];

    const int tid  = threadIdx.x;
    const int wave = tid >> 5;
    const int lane = tid & 31;
    const int half = lane >> 4;      // 0: lanes 0-15, 1: lanes 16-31
    const int lm   = lane & 15;
    const int rt0  = blockIdx.x * RT;

    // ---- cooperative, fused staging of the A panel into LDS ----
    constexpr int NV = RT * 16 * FIN / 4;  // float4 chunks
    for (int v = tid; v < NV; v += 256) {
        const int elem = v * 4;
        const int r    = elem / FIN;
        const int kk   = elem % FIN;
        const int g    = rt0 * 16 + r;
        float4 x = make_float4(0.f, 0.f, 0.f, 0.f);
        float so = 0.f, si = 0.f;
        if (g < n) {
            x  = *(const float4*)(A + (size_t)g * FIN + kk);
            so = rs_out[g];
            if (PRE) si = rs_in[g];
        }
        float vals[4] = {x.x, x.y, x.z, x.w};
#pragma unroll
        for (int j = 0; j < 4; ++j) {
            float t = vals[j];
            if (PRE) t = lrelu(t * si + bias[kk + j]);
            vals[j] = t * so;
        }
        float* p = &lds[r * LDK + kk];
        p[0] = vals[0]; p[1] = vals[1]; p[2] = vals[2]; p[3] = vals[3];
    }
    __syncthreads();

    const int rt = wave / CT;
    const int ct = wave % CT;
    const float* aRow = &lds[(rt * 16 + lm) * LDK];
    const int col = ct * 16 + lm;

    v8f c = {};
#pragma unroll
    for (int k = 0; k < FIN; k += 4) {
        // A frag (ISA 16x4 layout): lanes 0-15 hold K=k,k+1; lanes 16-31 K=k+2,k+3
        v2f a = *(const v2f*)(aRow + k + 2 * half);
        // B frag (4x16, rows striped across lanes; hi half-lanes = rows +2)
        const float* wp = W + (size_t)(k + 2 * half) * FOUT + col;
        v2f b;
        b.x = wp[0];
        b.y = wp[FOUT];
        c = __builtin_amdgcn_wmma_f32_16x16x4_f32(false, a, false, b, (short)0, c,
                                                  false, false);
    }

    // C/D layout: VGPR j -> row j (lanes 0-15) / row j+8 (lanes 16-31)
    const int grow = (rt0 + rt) * 16 + half * 8;
#pragma unroll
    for (int j = 0; j < 8; ++j) {
        const int r = grow + j;
        if (r < n) H[(size_t)r * FOUT + col] = c[j];
    }
}

// ---------------------------------------------------------------------------
// Edge scatter: one wave per edge moves a contiguous F-float feature row
// (coalesced 128B loads) into agg[dst] via global_atomic_add_f32.
// ---------------------------------------------------------------------------
template <int F>
__global__ void __launch_bounds__(256) edge_scatter(
        const float* __restrict__ H, const int* __restrict__ src,
        const int* __restrict__ dst, float* __restrict__ agg, int e) {
    constexpr int VW = F / 32;  // floats per lane (4 for F=128, 2 for F=64)
    const int w    = (blockIdx.x * 256 + threadIdx.x) >> 5;
    const int lane = threadIdx.x & 31;
    if (w >= e) return;
    const int s = src[w];
    const int d = dst[w];
    const float* hp = H + (size_t)s * F + lane * VW;
    float* ap       = agg + (size_t)d * F + lane * VW;
    float v[VW];
#pragma unroll
    for (int j = 0; j < VW; ++j) v[j] = hp[j];
#pragma unroll
    for (int j = 0; j < VW; ++j) atomicAdd(&ap[j], v[j]);
}

// ---------------------------------------------------------------------------
// Final epilogue (layer 2): out = lrelu(agg * rs_in[row] + b[col])
// ---------------------------------------------------------------------------
template <int F>
__global__ void finalize_out(const float* __restrict__ agg, const float* __restrict__ rs_in,
                             const float* __restrict__ bias, float* __restrict__ out, int n) {
    const int i = blockIdx.x * blockDim.x + threadIdx.x;
    if (i < n * F) {
        const int r = i / F;
        const int f = i % F;
        out[i] = lrelu(agg[i] * rs_in[r] + bias[f]);
    }
}

// ---------------------------------------------------------------------------
extern "C" void kernel_launch(void* const* d_in, const int* in_sizes, int n_in,
                              void* d_out, int out_size, void* d_ws, size_t ws_size,
                              hipStream_t stream) {
    const float* x  = (const float*)d_in[0];
    const int*   src = (const int*)d_in[1];
    const int*   dst = (const int*)d_in[2];
    const float* W0 = (const float*)d_in[3];
    const float* b0 = (const float*)d_in[4];
    const float* W1 = (const float*)d_in[5];
    const float* b1 = (const float*)d_in[6];
    const float* W2 = (const float*)d_in[7];
    const float* b2 = (const float*)d_in[8];

    const int n = in_sizes[0] / 128;
    const int e = in_sizes[1];

    float* ws     = (float*)d_ws;
    float* rs_out = ws;
    float* rs_in  = ws + (size_t)n;
    float* bufA   = ws + 2 * (size_t)n;              // GEMM output h   [n x 128]
    float* bufB   = bufA + (size_t)n * 128;          // scatter agg     [n x 128]

    // Degrees -> rsqrt (zero first; workspace is poisoned)
    hipMemsetAsync(rs_out, 0, 2 * (size_t)n * sizeof(float), stream);
    deg_accum<<<(e + 255) / 256, 256, 0, stream>>>(src, dst, rs_out, rs_in, e);
    deg_to_rsqrt<<<(n + 255) / 256, 256, 0, stream>>>(rs_out, rs_in, n);

    const int rowTiles = (n + 15) / 16;

    // ---- Layer 0: 128 -> 128 ----
    gcn_gemm_wmma<128, 128, false><<<rowTiles, 256, 0, stream>>>(
        x, rs_out, nullptr, nullptr, W0, bufA, n);
    hipMemsetAsync(bufB, 0, (size_t)n * 128 * sizeof(float), stream);
    edge_scatter<128><<<(e + 7) / 8, 256, 0, stream>>>(bufA, src, dst, bufB, e);

    // ---- Layer 1: 128 -> 64 (activation of layer 0 fused into A staging) ----
    gcn_gemm_wmma<128, 64, true><<<(rowTiles + 1) / 2, 256, 0, stream>>>(
        bufB, rs_out, rs_in, b0, W1, bufA, n);
    hipMemsetAsync(bufB, 0, (size_t)n * 64 * sizeof(float), stream);
    edge_scatter<64><<<(e + 7) / 8, 256, 0, stream>>>(bufA, src, dst, bufB, e);

    // ---- Layer 2: 64 -> 64 ----
    gcn_gemm_wmma<64, 64, true><<<(rowTiles + 1) / 2, 256, 0, stream>>>(
        bufB, rs_out, rs_in, b1, W2, bufA, n);
    hipMemsetAsync(bufB, 0, (size_t)n * 64 * sizeof(float), stream);
    edge_scatter<64><<<(e + 7) / 8, 256, 0, stream>>>(bufA, src, dst, bufB, e);

    // ---- Output epilogue ----
    const long long total = (long long)n * 64;
    finalize_out<64><<<(int)((total + 255) / 256), 256, 0, stream>>>(
        bufB, rs_in, b2, (float*)d_out, n);
}